// Xformer_58548994179677
// MI455X (gfx1250) — compile-verified
//
#include <hip/hip_runtime.h>

// Linformer attention pipeline for MI455X (gfx1250), all matmuls on
// v_wmma_f32_16x16x4_f32 (native fp32 matrix path, matches reference precision).

typedef __attribute__((ext_vector_type(2))) float v2f;
typedef __attribute__((ext_vector_type(8))) float v8f;

#define B_ 4
#define L_ 2046
#define D_ 1024
#define C_ 512      // D/S
#define P_ 1024     // L/S + 1
#define H_ 16
#define HD_ 32
#define DV_ 64
#define WKV_LD 2048 // w_kv row length (MAX_LEN)

static __device__ __forceinline__ v8f wmma4(v2f a, v2f b, v8f c) {
  // D = A(16x4,f32) * B(4x16,f32) + C(16x16,f32)
  return __builtin_amdgcn_wmma_f32_16x16x4_f32(false, a, false, b, (short)0, c,
                                               false, false);
}

static __device__ __forceinline__ float rowmax16(float v) {
  v = fmaxf(v, __shfl_xor(v, 8, 16));
  v = fmaxf(v, __shfl_xor(v, 4, 16));
  v = fmaxf(v, __shfl_xor(v, 2, 16));
  v = fmaxf(v, __shfl_xor(v, 1, 16));
  return v;
}
static __device__ __forceinline__ float rowsum16(float v) {
  v += __shfl_xor(v, 8, 16);
  v += __shfl_xor(v, 4, 16);
  v += __shfl_xor(v, 2, 16);
  v += __shfl_xor(v, 1, 16);
  return v;
}

// C = A @ W^T + bias, A row-major [M,K] (row stride strideAm), W row-major [N,K].
// One wave -> 32x64 output block (2 M-tiles x 4 N-tiles): each B-frag feeds
// 2 WMMAs -> 6 b64 loads : 8 wmma per k-step. A rows clamped (not masked) so
// the k-loop has no divergent branches; out-of-range rows only pollute
// accumulator rows that are never stored.
__global__ __launch_bounds__(256) void gemm_rowA_wmma(
    const float* __restrict__ A, long strideAm, long batchStrideA,
    const float* __restrict__ W, const float* __restrict__ bias,
    float* __restrict__ Out, long strideOm, long batchStrideO,
    int M, int K, int mstrips, int nstrips) {
  int lane = threadIdx.x & 31;
  int wid = (blockIdx.x * blockDim.x + threadIdx.x) >> 5;
  int wavesPerBatch = mstrips * nstrips;
  int b = wid / wavesPerBatch;
  int r = wid - b * wavesPerBatch;
  int m0 = (r / nstrips) * 32;
  int n0 = (r % nstrips) * 64;
  const float* Ab = A + (long)b * batchStrideA;
  float* Ob = Out + (long)b * batchStrideO;
  int col = lane & 15;
  int hi = lane >> 4;
  int khalf = hi * 2;
  int row0 = m0 + col;      if (row0 > M - 1) row0 = M - 1;
  int row1 = m0 + 16 + col; if (row1 > M - 1) row1 = M - 1;
  const float* Arow0 = Ab + (long)row0 * strideAm + khalf;
  const float* Arow1 = Ab + (long)row1 * strideAm + khalf;
  const float* Wp = W + (long)(n0 + col) * K + khalf;
  const long Wn16 = (long)16 * K;

  v8f acc[2][4] = {};
  for (int k0 = 0; k0 < K; k0 += 4) {
    if ((k0 & 63) == 0) {  // uniform branch: prefetch streams ~1KB ahead
      __builtin_prefetch(Arow0 + k0 + 256, 0, 1);
      __builtin_prefetch(Arow1 + k0 + 256, 0, 1);
      __builtin_prefetch(Wp + k0 + 256, 0, 1);
      __builtin_prefetch(Wp + 3 * Wn16 + k0 + 256, 0, 1);
    }
    v2f a0 = *(const v2f*)(Arow0 + k0);  // K-pair contiguous
    v2f a1 = *(const v2f*)(Arow1 + k0);
#pragma unroll
    for (int j = 0; j < 4; ++j) {
      // B[k][n] = W[n*K + k]; K-pair contiguous in a W row -> float2 load
      v2f bf = *(const v2f*)(Wp + j * Wn16 + k0);
      acc[0][j] = wmma4(a0, bf, acc[0][j]);
      acc[1][j] = wmma4(a1, bf, acc[1][j]);
    }
  }
  float bv[4];
#pragma unroll
  for (int j = 0; j < 4; ++j) bv[j] = bias[n0 + j * 16 + col];
#pragma unroll
  for (int i = 0; i < 2; ++i) {
#pragma unroll
    for (int rr = 0; rr < 8; ++rr) {
      int row = m0 + i * 16 + rr + 8 * hi;  // C frag row mapping
      if (row < M) {
        float* orow = Ob + (long)row * strideOm + n0;
#pragma unroll
        for (int j = 0; j < 4; ++j) orow[j * 16 + col] = acc[i][j][rr] + bv[j];
      }
    }
  }
}

// C = (seq-major A) @ Wp: A[m,k] = Ab[k*strideAk + m], B[k,n] = w_kv[n*2048+k].
// Out[m*strideOm + n]. M is a multiple of 32 here; K = L = 2046 -> one peeled
// guarded tail step, main loop branch-free.
__global__ __launch_bounds__(256) void gemm_colA_wmma(
    const float* __restrict__ A, long strideAk, long batchStrideA,
    const float* __restrict__ Wkv,
    float* __restrict__ Out, long strideOm, long batchStrideO,
    int K, int mstrips, int nstrips) {
  int lane = threadIdx.x & 31;
  int wid = (blockIdx.x * blockDim.x + threadIdx.x) >> 5;
  int wavesPerBatch = mstrips * nstrips;
  int b = wid / wavesPerBatch;
  int r = wid - b * wavesPerBatch;
  int m0 = (r / nstrips) * 32;
  int n0 = (r % nstrips) * 64;
  const float* Ab = A + (long)b * batchStrideA;
  float* Ob = Out + (long)b * batchStrideO;
  int col = lane & 15;
  int hi = lane >> 4;
  int khalf = hi * 2;
  const float* Am0 = Ab + (long)khalf * strideAk + m0 + col;       // rows m0..
  const float* Am1 = Am0 + 16;                                     // rows m0+16..
  const float* Wp = Wkv + (long)(n0 + col) * WKV_LD + khalf;
  const long Wn16 = (long)16 * WKV_LD;

  v8f acc[2][4] = {};
  int Kmain = K & ~3;
  for (int k0 = 0; k0 < Kmain; k0 += 4) {
    long ao = (long)k0 * strideAk;
    v2f a0 = {Am0[ao], Am0[ao + strideAk]};  // k, k+1 (row-strided operand)
    v2f a1 = {Am1[ao], Am1[ao + strideAk]};
#pragma unroll
    for (int j = 0; j < 4; ++j) {
      v2f bf = *(const v2f*)(Wp + j * Wn16 + k0);  // K-pair contiguous
      acc[0][j] = wmma4(a0, bf, acc[0][j]);
      acc[1][j] = wmma4(a1, bf, acc[1][j]);
    }
  }
  if (Kmain < K) {  // tail: K%4 = 2 -> upper K-half lanes contribute zero
    int k0 = Kmain;
    long ao = (long)k0 * strideAk;
    v2f a0 = {0.f, 0.f}, a1 = {0.f, 0.f};
    if (hi == 0) {
      a0 = (v2f){Am0[ao], Am0[ao + strideAk]};
      a1 = (v2f){Am1[ao], Am1[ao + strideAk]};
    }
#pragma unroll
    for (int j = 0; j < 4; ++j) {
      v2f bf = *(const v2f*)(Wp + j * Wn16 + k0);  // row len 2048: in-bounds
      acc[0][j] = wmma4(a0, bf, acc[0][j]);
      acc[1][j] = wmma4(a1, bf, acc[1][j]);
    }
  }
#pragma unroll
  for (int i = 0; i < 2; ++i) {
#pragma unroll
    for (int rr = 0; rr < 8; ++rr) {
      int row = m0 + i * 16 + rr + 8 * hi;  // always valid (32 | M)
      float* orow = Ob + (long)row * strideOm + n0;
#pragma unroll
      for (int j = 0; j < 4; ++j) orow[j * 16 + col] = acc[i][j][rr];
    }
  }
}

// Fused flash attention per (b, h, 16-row l-tile). One wave per tile.
// scores(16x16) = Qtile(16x32) @ Ktile(32x16); online softmax (exp2 domain);
// O(16x64) += P @ Vtile(16x64). Never materializes the 536MB attn matrix.
__global__ __launch_bounds__(256) void attn_wmma(
    const float* __restrict__ q, const float* __restrict__ kbuf,
    const float* __restrict__ vbuf, float* __restrict__ obuf) {
  __shared__ float lds[8][256];  // per-wave 16x16 prob tile for frag re-layout
  int lane = threadIdx.x & 31;
  int wslot = threadIdx.x >> 5;
  int wid = (blockIdx.x * blockDim.x + threadIdx.x) >> 5;
  int ltile = wid & 127;
  int h = (wid >> 7) & 15;
  int b = wid >> 11;
  int l0 = ltile * 16;
  int col = lane & 15;
  int hi = lane >> 4;
  int khalf = hi * 2;

  // Q A-frags: 8 K-steps over HEAD_DIM=32; fold scaling * log2(e) for exp2 softmax
  v2f qA[8];
  {
    int l = l0 + col;
    bool lv = l < L_;
    const float* qrow = q + ((long)b * L_ + l) * C_ + h * HD_;  // qh channel = h*32+d
    const float sc = 0.125f * 1.44269504088896340736f;          // SCALING * log2e
#pragma unroll
    for (int kk = 0; kk < 8; ++kk) {
      int k = kk * 4 + khalf;
      float x0 = lv ? qrow[k] * sc : 0.f;
      float x1 = lv ? qrow[k + 1] * sc : 0.f;
      qA[kk] = (v2f){x0, x1};
    }
  }

  v8f O[4] = {};
  float Mrow[8], Srow[8];
#pragma unroll
  for (int r = 0; r < 8; ++r) { Mrow[r] = -INFINITY; Srow[r] = 0.f; }

  const float* kb = kbuf + (long)b * C_ * P_ + (long)h * P_;  // kh channel = d*16+h
  const float* vb = vbuf + (long)b * D_ * P_ + (long)h * P_;  // vh channel = d*16+h

  for (int p0 = 0; p0 < P_; p0 += 16) {
    // ---- scores tile ----
    v8f c = {};
#pragma unroll
    for (int kk = 0; kk < 8; ++kk) {
      int k = kk * 4 + khalf;  // head-dim index
      const float* kp = kb + (long)k * 16 * P_ + p0 + col;
      v2f bf = {kp[0], kp[16 * P_]};  // channels d and d+1 -> stride 16*P
      c = wmma4(qA[kk], bf, c);
    }
    // ---- online softmax (rows split: lanes0-15 = rows r, lanes16-31 = rows r+8) ----
    float alpha[8];
#pragma unroll
    for (int r = 0; r < 8; ++r) {
      float mx = rowmax16(c[r]);
      float mnew = fmaxf(Mrow[r], mx);
      float al = __builtin_exp2f(Mrow[r] - mnew);
      float pv = __builtin_exp2f(c[r] - mnew);
      Srow[r] = Srow[r] * al + rowsum16(pv);
      Mrow[r] = mnew;
      alpha[r] = al;
      lds[wslot][(r + 8 * hi) * 16 + col] = pv;  // C-frag layout -> LDS
    }
#pragma unroll
    for (int r = 0; r < 8; ++r) {
      O[0][r] *= alpha[r]; O[1][r] *= alpha[r];
      O[2][r] *= alpha[r]; O[3][r] *= alpha[r];
    }
    __builtin_amdgcn_fence(__ATOMIC_ACQ_REL, "wavefront");
    __builtin_amdgcn_wave_barrier();
    // re-read probs as A-frags (row = lane&15, K-pair = p offset)
    v2f pA[4];
#pragma unroll
    for (int kk = 0; kk < 4; ++kk) {
      int k = kk * 4 + khalf;
      pA[kk] = (v2f){lds[wslot][col * 16 + k], lds[wslot][col * 16 + k + 1]};
    }
    __builtin_amdgcn_fence(__ATOMIC_ACQ_REL, "wavefront");
    __builtin_amdgcn_wave_barrier();
    // ---- O += P @ V ----
#pragma unroll
    for (int j = 0; j < 4; ++j) {
#pragma unroll
      for (int kk = 0; kk < 4; ++kk) {
        int k = kk * 4 + khalf;  // p offset in tile
        const float* vp = vb + (long)((j * 16 + col) * 16) * P_ + p0 + k;
        v2f bf = *(const v2f*)vp;  // p, p+1 contiguous
        O[j] = wmma4(pA[kk], bf, O[j]);
      }
    }
  }

  // normalize + store to pre-projection layout obuf[b, l, d*16+h]
#pragma unroll
  for (int r = 0; r < 8; ++r) {
    float inv = 1.f / Srow[r];
    O[0][r] *= inv; O[1][r] *= inv; O[2][r] *= inv; O[3][r] *= inv;
  }
#pragma unroll
  for (int r = 0; r < 8; ++r) {
    int l = l0 + r + 8 * hi;
    if (l < L_) {
      float* orow = obuf + ((long)b * L_ + l) * D_ + h;
#pragma unroll
      for (int j = 0; j < 4; ++j) orow[(j * 16 + col) * 16] = O[j][r];
    }
  }
}

extern "C" void kernel_launch(void* const* d_in, const int* in_sizes, int n_in,
                              void* d_out, int out_size, void* d_ws,
                              size_t ws_size, hipStream_t stream) {
  (void)in_sizes; (void)n_in; (void)out_size; (void)ws_size;
  const float* in = (const float*)d_in[0];     // [L, B, D]
  const float* w_kv = (const float*)d_in[1];   // [1024, 2048]
  const float* w_q = (const float*)d_in[2];    // [512, 1024]
  const float* b_q = (const float*)d_in[3];    // [512]
  const float* w_out = (const float*)d_in[4];  // [1024, 1024]
  const float* b_out = (const float*)d_in[5];  // [1024]
  float* out = (float*)d_out;                  // [L, B, D]

  float* ws = (float*)d_ws;  // ~75.4 MB of fp32 scratch (fits easily in L2)
  float* qbuf = ws;                                    // [B, L, C]
  float* vbuf = qbuf + (size_t)B_ * L_ * C_;           // [B, D, P]
  float* kbuf = vbuf + (size_t)B_ * D_ * P_;           // [B, C, P]
  float* obuf = kbuf + (size_t)B_ * C_ * P_;           // [B, L, D]

  // 1) q = X @ w_q.T + b_q : A[m,k]=in[m*B*D + b*D + k]
  //    waves = 4 * ceil(2046/32)=64 * (512/64)=8 = 2048 -> 256 blocks
  gemm_rowA_wmma<<<256, 256, 0, stream>>>(in, (long)(B_ * D_), (long)D_, w_q,
                                          b_q, qbuf, (long)C_, (long)L_ * C_,
                                          L_, D_, 64, 8);
  // 2a) v[b,d,p] = sum_l in[l,b,d]*w_kv[p,l] : A[m,k]=in[k*B*D + b*D + m]
  //    waves = 4 * 32 * 16 = 2048 -> 256 blocks
  gemm_colA_wmma<<<256, 256, 0, stream>>>(in, (long)(B_ * D_), (long)D_, w_kv,
                                          vbuf, (long)P_, (long)D_ * P_, L_,
                                          32, 16);
  // 2b) k[b,c,p] = sum_l q[b,l,c]*w_kv[p,l] : A[m,k]=qbuf[b*L*C + k*C + m]
  //    waves = 4 * 16 * 16 = 1024 -> 128 blocks
  gemm_colA_wmma<<<128, 256, 0, stream>>>(qbuf, (long)C_, (long)L_ * C_, w_kv,
                                          kbuf, (long)P_, (long)C_ * P_, L_,
                                          16, 16);
  // 3) fused flash attention -> obuf[b,l,(d*16+h)]
  //    waves = 4*16*128 = 8192 -> 1024 blocks
  attn_wmma<<<1024, 256, 0, stream>>>(qbuf, kbuf, vbuf, obuf);
  // 4) out = O @ w_out.T + b_out -> d_out[l, b, :]
  //    waves = 4 * 64 * 16 = 4096 -> 512 blocks
  gemm_rowA_wmma<<<512, 256, 0, stream>>>(obuf, (long)D_, (long)L_ * D_,
                                          w_out, b_out, out, (long)(B_ * D_),
                                          (long)D_, L_, D_, 64, 16);
}